// GCN_single_output_8280696947375
// MI455X (gfx1250) — compile-verified
//
#include <hip/hip_runtime.h>
#include <hip/hip_bf16.h>

// ---------------------------------------------------------------------------
// GCN 2-layer forward for MI455X (gfx1250), wave32.
//   layer1: h = x@W1 ; agg = scatter(norm * h[src] -> dst) + selfloop ; relu(+b1)
//   layer2: z = h2@W2 ; out = scatter(norm * z[src] -> dst) + selfloop + b2
// Dense 3->16 transform uses V_WMMA_F32_16X16X4_F32 (one 16x16 f32 tile per
// wave, K=4 with K=3 zero-padded). Everything else is L2-resident scatter.
// ---------------------------------------------------------------------------

typedef __attribute__((ext_vector_type(2))) float v2f;
typedef __attribute__((ext_vector_type(8))) float v8f;

// ---- init: deg = 1.0 (self-loop), agg1 = 0 --------------------------------
__global__ void init_kernel(float* __restrict__ deg, float* __restrict__ agg1,
                            int N) {
  int i = blockIdx.x * blockDim.x + threadIdx.x;
  if (i < N * 16) agg1[i] = 0.0f;
  if (i < N) deg[i] = 1.0f;
}

// ---- degree: deg[dst] += 1 ------------------------------------------------
__global__ void degree_kernel(const int* __restrict__ dst,
                              float* __restrict__ deg, int E) {
  int e = blockIdx.x * blockDim.x + threadIdx.x;
  if (e >= E) return;
  atomicAdd(&deg[dst[e]], 1.0f);
}

// ---- dinv = rsqrt(deg) in place (deg >= 1 always, self-loop) --------------
__global__ void rsqrt_kernel(float* __restrict__ deg, int N) {
  int i = blockIdx.x * blockDim.x + threadIdx.x;
  if (i >= N) return;
  float d = deg[i];
  deg[i] = (d > 0.0f) ? __frsqrt_rn(d) : 0.0f;
}

// ---- h = x @ W1 via V_WMMA_F32_16X16X4_F32 --------------------------------
// A 16x4 f32 layout : lanes 0-15 -> M=lane, vgpr{0,1}=K{0,1};
//                     lanes 16-31 -> M=lane-16, vgpr{0,1}=K{2,3}
// B 4x16  f32 layout: vgpr0 -> K=0 (lanes 0-15) / K=2 (lanes 16-31);
//                     vgpr1 -> K=1 / K=3 ; N = lane%16
// D 16x16 f32 layout: vgpr v -> row v (lanes 0-15) / row v+8 (lanes 16-31)
__global__ void __launch_bounds__(320)
gemm1_wmma(const float* __restrict__ x,   // [N][3]
           const float* __restrict__ W1,  // [3][16]
           float* __restrict__ h,         // [N][16]
           int num_tiles) {
  int flat = blockIdx.x * blockDim.x + threadIdx.x;
  int wave = flat >> 5;
  int lane = threadIdx.x & 31;
  if (wave >= num_tiles) return;  // wave-uniform: EXEC stays all-1s for WMMA

  int m = lane & 15;
  bool hi = lane >= 16;
  int row0 = wave * 16;

  const float* xr = x + (size_t)(row0 + m) * 3;
  v2f a;
  a.x = hi ? xr[2] : xr[0];
  a.y = hi ? 0.0f : xr[1];   // K=3 pad / K=1

  v2f b;
  b.x = hi ? W1[2 * 16 + m] : W1[0 * 16 + m];
  b.y = hi ? 0.0f : W1[1 * 16 + m];  // K=3 pad / K=1

  v8f c = {};
  c = __builtin_amdgcn_wmma_f32_16x16x4_f32(
      /*neg_a=*/false, a, /*neg_b=*/false, b,
      /*c_mod=*/(short)0, c, /*reuse_a=*/false, /*reuse_b=*/false);

  int rbase = row0 + (hi ? 8 : 0);
#pragma unroll
  for (int v = 0; v < 8; ++v) {
    h[(size_t)(rbase + v) * 16 + m] = c[v];
  }
}

// ---- layer-1 edge scatter: agg1[dst] += norm * h[src] ---------------------
__global__ void edge_pass1(const int* __restrict__ src,
                           const int* __restrict__ dst,
                           const float* __restrict__ dinv,
                           const float4* __restrict__ h4,  // [N][4] of float4
                           float* __restrict__ agg1,       // [N][16]
                           int E) {
  int e = blockIdx.x * blockDim.x + threadIdx.x;
  if (e >= E) return;
  int s = src[e];
  int d = dst[e];
  float norm = dinv[s] * dinv[d];
  const float4* hs = h4 + (size_t)s * 4;
  float* out = agg1 + (size_t)d * 16;
#pragma unroll
  for (int g = 0; g < 4; ++g) {
    float4 v = hs[g];  // global_load_b128, L2-resident gather
    atomicAdd(out + 4 * g + 0, v.x * norm);
    atomicAdd(out + 4 * g + 1, v.y * norm);
    atomicAdd(out + 4 * g + 2, v.z * norm);
    atomicAdd(out + 4 * g + 3, v.w * norm);
  }
}

// ---- fuse: self-loop + b1 + relu, z = h2 @ W2, out init = z*dinv^2 + b2 ---
__global__ void finalize1_project(const float* __restrict__ h,     // x@W1
                                  const float* __restrict__ agg1,  // edges
                                  const float* __restrict__ dinv,
                                  const float* __restrict__ b1,   // [16]
                                  const float* __restrict__ W2,   // [16]
                                  const float* __restrict__ b2,   // [1]
                                  float* __restrict__ z,          // [N]
                                  float* __restrict__ out,        // d_out [N]
                                  int N) {
  int i = blockIdx.x * blockDim.x + threadIdx.x;
  if (i >= N) return;
  float di = dinv[i];
  float selfw = di * di;
  float acc = 0.0f;
  const float4* a4 = (const float4*)(agg1 + (size_t)i * 16);
  const float4* h4 = (const float4*)(h + (size_t)i * 16);
#pragma unroll
  for (int g = 0; g < 4; ++g) {
    float4 av = a4[g];
    float4 hv = h4[g];
    float v0 = fmaxf(av.x + hv.x * selfw + b1[4 * g + 0], 0.0f);
    float v1 = fmaxf(av.y + hv.y * selfw + b1[4 * g + 1], 0.0f);
    float v2 = fmaxf(av.z + hv.z * selfw + b1[4 * g + 2], 0.0f);
    float v3 = fmaxf(av.w + hv.w * selfw + b1[4 * g + 3], 0.0f);
    acc += v0 * W2[4 * g + 0] + v1 * W2[4 * g + 1] +
           v2 * W2[4 * g + 2] + v3 * W2[4 * g + 3];
  }
  z[i] = acc;
  out[i] = acc * selfw + b2[0];  // layer-2 self-loop + bias; full overwrite
}

// ---- layer-2 edge scatter: out[dst] += norm * z[src] ----------------------
__global__ void edge_pass2(const int* __restrict__ src,
                           const int* __restrict__ dst,
                           const float* __restrict__ dinv,
                           const float* __restrict__ z,
                           float* __restrict__ out, int E) {
  int e = blockIdx.x * blockDim.x + threadIdx.x;
  if (e >= E) return;
  int s = src[e];
  int d = dst[e];
  atomicAdd(&out[d], z[s] * dinv[s] * dinv[d]);
}

extern "C" void kernel_launch(void* const* d_in, const int* in_sizes, int n_in,
                              void* d_out, int out_size, void* d_ws,
                              size_t ws_size, hipStream_t stream) {
  const float* x  = (const float*)d_in[0];        // [N][3]
  const int*   ei = (const int*)d_in[1];          // [2][E]
  const float* W1 = (const float*)d_in[2];        // [3][16]
  const float* b1 = (const float*)d_in[3];        // [16]
  const float* W2 = (const float*)d_in[4];        // [16][1]
  const float* b2 = (const float*)d_in[5];        // [1]
  float* out = (float*)d_out;

  const int N = in_sizes[0] / 3;
  const int E = in_sizes[1] / 2;
  const int* src = ei;
  const int* dst = ei + E;

  // Workspace layout (all offsets 16B-aligned; N*4 = 800000 bytes).
  char* ws = (char*)d_ws;
  float* dinv = (float*)(ws);                                   // N
  float* h    = (float*)(ws + (size_t)N * 4);                   // N*16
  float* agg1 = (float*)(ws + (size_t)N * 4 + (size_t)N * 64);  // N*16
  float* z    = (float*)(ws + (size_t)N * 4 + (size_t)N * 128); // N

  const int BT = 256;

  // 1) deg=1 (self-loop), agg1=0
  init_kernel<<<(N * 16 + BT - 1) / BT, BT, 0, stream>>>(dinv, agg1, N);
  // 2) deg[dst] += 1
  degree_kernel<<<(E + BT - 1) / BT, BT, 0, stream>>>(dst, dinv, E);
  // 3) dinv = rsqrt(deg)
  rsqrt_kernel<<<(N + BT - 1) / BT, BT, 0, stream>>>(dinv, N);
  // 4) h = x @ W1 via f32 WMMA, one 16x16 tile per wave (N % 16 == 0)
  {
    int ntiles = N / 16;
    int wpb = 10;                       // 320 threads = 10 waves
    int blocks = (ntiles + wpb - 1) / wpb;
    gemm1_wmma<<<blocks, 32 * wpb, 0, stream>>>(x, W1, h, ntiles);
  }
  // 5) layer-1 edge scatter
  edge_pass1<<<(E + BT - 1) / BT, BT, 0, stream>>>(src, dst, dinv,
                                                   (const float4*)h, agg1, E);
  // 6) self-loop + bias + relu + project to z, initialize d_out
  finalize1_project<<<(N + BT - 1) / BT, BT, 0, stream>>>(h, agg1, dinv, b1,
                                                          W2, b2, z, out, N);
  // 7) layer-2 edge scatter
  edge_pass2<<<(E + BT - 1) / BT, BT, 0, stream>>>(src, dst, dinv, z, out, E);
}